// Net_28432683499602
// MI455X (gfx1250) — compile-verified
//
#include <hip/hip_runtime.h>
#include <hip/hip_bf16.h>
#include <math.h>

// ---------------------------------------------------------------------------
// SplineConv (K=2, degree-1 B-spline, mean agg), 2 layers + log_softmax.
// MI455X plan:
//   * Phase 1 (dominant): y = x @ [W1_0 | W1_1 | root1]; x = 573MB f32 stream
//     -> bandwidth bound (~25us @ 23.3TB/s). V_WMMA_F32_16X16X4_F32, 16-row
//     tile per wave32, 3 f32 accumulators. B (275KB) is staged per-workgroup
//     into LDS in 44.5KB K-chunks with a pair-interleaved layout so each
//     K-step needs just 3 conflict-free ds_load_b64 ops; VMEM port is left
//     entirely to the x stream. 10 waves/block, 625 blocks = 6250 tiles exact
//     (no divergence around barriers, EXEC full for every WMMA).
//   * Phase 2: per-edge interpolate + global_atomic_add_f32 scatter into a
//     6.4MB accumulator (L2-resident on the 192MB L2).
//   * Phase 3: layer-2 WMMA GEMM (K=16 unrolled), edge scatter (7 cols),
//     then mean + bias + log_softmax.
// ---------------------------------------------------------------------------

typedef __attribute__((ext_vector_type(2))) float v2f;
typedef __attribute__((ext_vector_type(8))) float v8f;

#define NN    100000
#define NE    3200000
#define INF_  1433
#define KMAIN 1432              // multiple-of-4 part of K; tail handled solo
#define KC    232               // K-chunk staged in LDS (multiple of 4)
#define HID   16
#define OUTF  7
#define LDSM  (KC * HID)        // floats per B matrix in LDS (3712)

static __device__ __forceinline__ v8f wmma_f32_k4(v2f a, v2f b, v8f c) {
  // D = A(16x4,f32) * B(4x16,f32) + C(16x16,f32)
  return __builtin_amdgcn_wmma_f32_16x16x4_f32(
      /*neg_a=*/false, a, /*neg_b=*/false, b,
      /*c_mod=*/(short)0, c, /*reuse_a=*/false, /*reuse_b=*/false);
}

// ---- Layer 1 fused GEMM: [NN x 1433] @ three [1433 x 16] -> three [NN x 16]
// 320 threads = 10 waves, 1 tile each; 625 blocks * 10 = 6250 = NN/16 exactly.
__global__ __launch_bounds__(320) void spl_gemm1(
    const float* __restrict__ x,
    const float* __restrict__ w0, const float* __restrict__ w1,
    const float* __restrict__ wr,
    float* __restrict__ xw0, float* __restrict__ xw1, float* __restrict__ xr) {
  __shared__ float ldsB[3 * LDSM];             // 44544 bytes

  const int lane = threadIdx.x & 31;
  const int wave = threadIdx.x >> 5;
  const int tile = blockIdx.x * 10 + wave;     // always < 6250
  const int row0 = tile * 16;
  const int m  = lane & 15;                    // A: row in tile; B/D: column
  const int hi = lane >> 4;                    // K half-select per ISA layout
  const float* xrow = x + (size_t)(row0 + m) * INF_;

  v8f c0 = {}; v8f c1 = {}; v8f c2 = {};

  for (int kc = 0; kc < KMAIN; kc += KC) {
    const int kcLen = (KMAIN - kc) < KC ? (KMAIN - kc) : KC;  // 232 or 40
    __syncthreads();
    // Stage B chunk: pair-interleave so (k,k+1)@col m is one b64 in LDS.
    for (int idx = threadIdx.x; idx < kcLen * HID; idx += 320) {
      const int kL = idx >> 4, mm = idx & 15;
      const int g  = (kc + kL) * HID + mm;                 // coalesced global
      const int l  = ((kL >> 1) * HID + mm) * 2 + (kL & 1);
      ldsB[l]            = w0[g];
      ldsB[l + LDSM]     = w1[g];
      ldsB[l + 2 * LDSM] = wr[g];
    }
    __syncthreads();
    for (int k = 0; k < kcLen; k += 4) {
      const int kk = kc + k + hi * 2;          // lanes0-15: k,k+1; 16-31: +2,+3
      v2f a; a.x = xrow[kk]; a.y = xrow[kk + 1];
      const int lb = ((k >> 1) + hi) * (HID * 2) + m * 2;  // 8B-aligned
      v2f b0 = *(const v2f*)&ldsB[lb];
      v2f b1 = *(const v2f*)&ldsB[lb + LDSM];
      v2f b2 = *(const v2f*)&ldsB[lb + 2 * LDSM];
      c0 = wmma_f32_k4(a, b0, c0);
      c1 = wmma_f32_k4(a, b1, c1);
      c2 = wmma_f32_k4(a, b2, c2);
    }
  }
  { // K tail: 1433 % 4 == 1 -> one valid column, only hi==0 lanes' .x slot
    v2f a = {0.f, 0.f}, b0 = {0.f, 0.f}, b1 = {0.f, 0.f}, b2 = {0.f, 0.f};
    if (hi == 0) {
      a.x  = xrow[INF_ - 1];
      b0.x = w0[(INF_ - 1) * HID + m];
      b1.x = w1[(INF_ - 1) * HID + m];
      b2.x = wr[(INF_ - 1) * HID + m];
    }
    c0 = wmma_f32_k4(a, b0, c0);               // EXEC is full again here
    c1 = wmma_f32_k4(a, b1, c1);
    c2 = wmma_f32_k4(a, b2, c2);
  }
  // D layout: vgpr i, lane -> (M = i + 8*hi, N = m); coalesced 64B stores
  const size_t ob = (size_t)(row0 + hi * 8) * HID + m;
#pragma unroll
  for (int i = 0; i < 8; ++i) {
    xw0[ob + (size_t)i * HID] = c0[i];
    xw1[ob + (size_t)i * HID] = c1[i];
    xr [ob + (size_t)i * HID] = c2[i];
  }
}

// ---- Layer 1 edge scatter: one lane per (edge, feature)
__global__ __launch_bounds__(256) void spl_edge1(
    const int* __restrict__ src, const int* __restrict__ dst,
    const float* __restrict__ pseudo,
    const float* __restrict__ xw0, const float* __restrict__ xw1,
    float* __restrict__ agg, float* __restrict__ deg) {
  const long long t = (long long)blockIdx.x * blockDim.x + threadIdx.x;
  const long long e = t >> 4;
  const int f = (int)(t & 15);
  if (e >= NE) return;
  const int s = src[e];
  const int d = dst[e];
  const float u = pseudo[e];
  const float msg = (1.0f - u) * xw0[(size_t)s * HID + f]
                  +         u  * xw1[(size_t)s * HID + f];
  atomicAdd(&agg[(size_t)d * HID + f], msg);
  if (f == 0) atomicAdd(&deg[d], 1.0f);
}

// ---- Layer 1 finalize: mean + root + bias, ELU
__global__ __launch_bounds__(256) void spl_fin1(
    const float* __restrict__ agg, const float* __restrict__ deg,
    const float* __restrict__ xr, const float* __restrict__ bias,
    float* __restrict__ h) {
  const long long t = (long long)blockIdx.x * blockDim.x + threadIdx.x;
  const long long n = t >> 4;
  const int f = (int)(t & 15);
  if (n >= NN) return;
  const float dg = fmaxf(deg[n], 1.0f);
  const float v = agg[(size_t)n * HID + f] / dg
                + xr[(size_t)n * HID + f] + bias[f];
  h[(size_t)n * HID + f] = (v > 0.0f) ? v : (__expf(v) - 1.0f);  // ELU
}

// ---- Layer 2 GEMM: [NN x 16] @ three [16 x 7] -> three [NN x 7]
__global__ __launch_bounds__(256) void spl_gemm2(
    const float* __restrict__ h,
    const float* __restrict__ w0, const float* __restrict__ w1,
    const float* __restrict__ wr,
    float* __restrict__ hw0, float* __restrict__ hw1, float* __restrict__ hr) {
  const int lane = threadIdx.x & 31;
  const int wave = threadIdx.x >> 5;
  const int tile = blockIdx.x * 8 + wave;
  if (tile >= NN / 16) return;                 // wave-uniform
  const int row0 = tile * 16;
  const int m  = lane & 15;
  const int hi = lane >> 4;
  const float* hrow = h + (size_t)(row0 + m) * HID;

  v8f c0 = {}; v8f c1 = {}; v8f c2 = {};
#pragma unroll
  for (int k = 0; k < HID; k += 4) {
    const int kk = k + hi * 2;
    v2f a;  a.x = hrow[kk]; a.y = hrow[kk + 1];
    v2f b0 = {0.f, 0.f}, b1 = {0.f, 0.f}, b2 = {0.f, 0.f};
    if (m < OUTF) {                             // N=7: pad columns with zero
      b0.x = w0[kk * OUTF + m]; b0.y = w0[(kk + 1) * OUTF + m];
      b1.x = w1[kk * OUTF + m]; b1.y = w1[(kk + 1) * OUTF + m];
      b2.x = wr[kk * OUTF + m]; b2.y = wr[(kk + 1) * OUTF + m];
    }
    c0 = wmma_f32_k4(a, b0, c0);                // EXEC full at the WMMA
    c1 = wmma_f32_k4(a, b1, c1);
    c2 = wmma_f32_k4(a, b2, c2);
  }
  if (m < OUTF) {
    const size_t ob = (size_t)(row0 + hi * 8) * OUTF + m;
#pragma unroll
    for (int i = 0; i < 8; ++i) {
      hw0[ob + (size_t)i * OUTF] = c0[i];
      hw1[ob + (size_t)i * OUTF] = c1[i];
      hr [ob + (size_t)i * OUTF] = c2[i];
    }
  }
}

// ---- Layer 2 edge scatter: 8 lanes per edge (7 active)
__global__ __launch_bounds__(256) void spl_edge2(
    const int* __restrict__ src, const int* __restrict__ dst,
    const float* __restrict__ pseudo,
    const float* __restrict__ hw0, const float* __restrict__ hw1,
    float* __restrict__ agg) {
  const long long t = (long long)blockIdx.x * blockDim.x + threadIdx.x;
  const long long e = t >> 3;
  const int j = (int)(t & 7);
  if (e >= NE || j >= OUTF) return;
  const int s = src[e];
  const int d = dst[e];
  const float u = pseudo[e];
  const float msg = (1.0f - u) * hw0[(size_t)s * OUTF + j]
                  +         u  * hw1[(size_t)s * OUTF + j];
  atomicAdd(&agg[(size_t)d * OUTF + j], msg);
}

// ---- Layer 2 finalize + log_softmax (one thread per node)
__global__ __launch_bounds__(256) void spl_fin2(
    const float* __restrict__ agg, const float* __restrict__ deg,
    const float* __restrict__ hr, const float* __restrict__ bias,
    float* __restrict__ out) {
  const long long n = (long long)blockIdx.x * blockDim.x + threadIdx.x;
  if (n >= NN) return;
  const float dg = fmaxf(deg[n], 1.0f);
  float o[OUTF];
  float mx = -3.402823466e38f;
#pragma unroll
  for (int j = 0; j < OUTF; ++j) {
    o[j] = agg[(size_t)n * OUTF + j] / dg + hr[(size_t)n * OUTF + j] + bias[j];
    mx = fmaxf(mx, o[j]);
  }
  float s = 0.0f;
#pragma unroll
  for (int j = 0; j < OUTF; ++j) s += __expf(o[j] - mx);
  const float lse = __logf(s) + mx;
#pragma unroll
  for (int j = 0; j < OUTF; ++j) out[(size_t)n * OUTF + j] = o[j] - lse;
}

extern "C" void kernel_launch(void* const* d_in, const int* in_sizes, int n_in,
                              void* d_out, int out_size, void* d_ws, size_t ws_size,
                              hipStream_t stream) {
  (void)in_sizes; (void)n_in; (void)out_size; (void)ws_size;
  const float* x     = (const float*)d_in[0];
  const int*   ei    = (const int*)d_in[1];           // [2][NE]
  const float* ps    = (const float*)d_in[2];         // [NE][1]
  const float* W1    = (const float*)d_in[3];         // [2][1433][16]
  const float* root1 = (const float*)d_in[4];         // [1433][16]
  const float* bias1 = (const float*)d_in[5];         // [16]
  const float* W2    = (const float*)d_in[6];         // [2][16][7]
  const float* root2 = (const float*)d_in[7];         // [16][7]
  const float* bias2 = (const float*)d_in[8];         // [7]
  float* out = (float*)d_out;

  const int* srcIdx = ei;
  const int* dstIdx = ei + NE;
  const float* w10 = W1;
  const float* w11 = W1 + (size_t)INF_ * HID;
  const float* w20 = W2;
  const float* w21 = W2 + (size_t)HID * OUTF;

  // Workspace carve-up (~43.6 MB of f32)
  float* ws   = (float*)d_ws;
  float* xw0  = ws;
  float* xw1  = xw0  + (size_t)NN * HID;
  float* xr   = xw1  + (size_t)NN * HID;
  float* agg1 = xr   + (size_t)NN * HID;
  float* deg  = agg1 + (size_t)NN * HID;   // adjacent to agg1 (joint memset)
  float* h    = deg  + (size_t)NN;
  float* hw0  = h    + (size_t)NN * HID;
  float* hw1  = hw0  + (size_t)NN * OUTF;
  float* hr   = hw1  + (size_t)NN * OUTF;
  float* agg2 = hr   + (size_t)NN * OUTF;

  // Zero accumulators every call (capture-safe async memset on stream)
  hipMemsetAsync(agg1, 0, (size_t)NN * (HID + 1) * sizeof(float), stream); // agg1+deg
  hipMemsetAsync(agg2, 0, (size_t)NN * OUTF * sizeof(float), stream);

  spl_gemm1<<<625, 320, 0, stream>>>(x, w10, w11, root1, xw0, xw1, xr);

  {
    const long long th = (long long)NE * 16;
    spl_edge1<<<(int)((th + 255) / 256), 256, 0, stream>>>(
        srcIdx, dstIdx, ps, xw0, xw1, agg1, deg);
  }
  {
    const long long th = (long long)NN * 16;
    spl_fin1<<<(int)((th + 255) / 256), 256, 0, stream>>>(
        agg1, deg, xr, bias1, h);
  }

  spl_gemm2<<<(NN / 16 + 7) / 8, 256, 0, stream>>>(h, w20, w21, root2, hw0, hw1, hr);

  {
    const long long th = (long long)NE * 8;
    spl_edge2<<<(int)((th + 255) / 256), 256, 0, stream>>>(
        srcIdx, dstIdx, ps, hw0, hw1, agg2);
  }
  spl_fin2<<<(NN + 255) / 256, 256, 0, stream>>>(agg2, deg, hr, bias2, out);
}